// HGCNEncoder_57698590654796
// MI455X (gfx1250) — compile-verified
//
#include <hip/hip_runtime.h>
#include <hip/hip_bf16.h>

// ---------------------------------------------------------------------------
// HGCN encoder for MI455X (gfx1250, wave32).
//   h = x @ W               -> V_WMMA_F32_16X16X4_F32, W staged in LDS
//   deg/deg_inv_sqrt        -> native f32 atomics into L2-resident 400KB array
//   out = scatter_add(norm * h[row]) + bias  -> global_atomic_add_f32
//                                               (unsafeAtomicAdd => native HW
//                                               atomic, no CAS loop), one
//                                               wave per edge, float2 gathers
// ---------------------------------------------------------------------------

typedef float v2f __attribute__((ext_vector_type(2)));
typedef float v8f __attribute__((ext_vector_type(8)));

#define IN_CH  256
#define OUT_CH 64

// ---- kernel 1: out = bias (broadcast), deg = 0 ----------------------------
__global__ __launch_bounds__(256)
void hgcn_init(float* __restrict__ out, const float* __restrict__ bias,
               float* __restrict__ deg, int N) {
    int i = blockIdx.x * 256 + threadIdx.x;
    int total = N * OUT_CH;
    if (i < total) out[i] = bias[i & (OUT_CH - 1)];
    if (i < N)     deg[i] = 0.0f;
}

// ---- kernel 2: deg[col] += 1 ----------------------------------------------
__global__ __launch_bounds__(256)
void hgcn_deg(const long long* __restrict__ ei, float* __restrict__ deg, int E) {
    int i = blockIdx.x * 256 + threadIdx.x;
    if (i < E) {
        long long col = ei[(size_t)E + i];
        unsafeAtomicAdd(&deg[col], 1.0f);   // native global_atomic_add_f32
    }
}

// ---- kernel 3: deg -> deg_inv_sqrt in place -------------------------------
__global__ __launch_bounds__(256)
void hgcn_dis(float* __restrict__ deg, int N) {
    int i = blockIdx.x * 256 + threadIdx.x;
    if (i < N) {
        float d = deg[i];
        deg[i] = (d > 0.0f) ? rsqrtf(fmaxf(d, 1.0f)) : 0.0f;
    }
}

// ---- kernel 4: h = x @ W via V_WMMA_F32_16X16X4_F32 -----------------------
// Block = 128 threads = 4 waves. Each wave owns one 16-row node tile and the
// full 64 output columns (4 accumulators of v8f). W (64KB) is cooperatively
// staged into LDS once per block. K loop: 256 in steps of 4.
//
// Fragment layouts (ISA 7.12.2, wave32):
//   A 16x4 f32 : lane L holds row M=L%16; vgpr j holds K = 2*(L/16) + j
//   B  4x16 f32: lane L holds col N=L%16; vgpr j holds K = 2*(L/16) + j
//   C/D 16x16  : vgpr r holds row M = r + 8*(L/16), col N = L%16
__global__ __launch_bounds__(128)
void hgcn_gemm_wmma(const float* __restrict__ x, const float* __restrict__ W,
                    float* __restrict__ h, int nTiles, int N) {
    __shared__ float Wlds[IN_CH * OUT_CH];   // 64 KB of the 320 KB/WGP

    // cooperative W load: 16384 floats / 128 threads, float4 = 32 iterations
    {
        const float4* Wv = (const float4*)W;
        float4*       Lv = (float4*)Wlds;
        for (int i = threadIdx.x; i < (IN_CH * OUT_CH) / 4; i += 128)
            Lv[i] = Wv[i];
    }
    __syncthreads();

    const int wave = threadIdx.x >> 5;
    const int lane = threadIdx.x & 31;
    int tile = blockIdx.x * 4 + wave;
    // Clamp instead of early-exit: keeps EXEC all-ones for every v_wmma
    // (tail waves redundantly recompute the last tile; stores are identical).
    if (tile >= nTiles) tile = nTiles - 1;

    const int m   = lane & 15;          // A row / C col within tile
    const int khi = (lane >> 4) << 1;   // 0 or 2: K sub-offset for this half
    int arow = tile * 16 + m;
    if (arow > N - 1) arow = N - 1;     // robustness for N % 16 != 0
    const float* aptr = x + (size_t)arow * IN_CH + khi;

    v8f acc0 = {}, acc1 = {}, acc2 = {}, acc3 = {};

    #pragma unroll 4
    for (int k0 = 0; k0 < IN_CH; k0 += 4) {
        v2f a = *(const v2f*)(aptr + k0);              // K = k0+khi, k0+khi+1
        const float* wb = &Wlds[(k0 + khi) * OUT_CH + m];

        v2f b0, b1, b2, b3;
        b0.x = wb[0];               b0.y = wb[OUT_CH + 0];
        b1.x = wb[16];              b1.y = wb[OUT_CH + 16];
        b2.x = wb[32];              b2.y = wb[OUT_CH + 32];
        b3.x = wb[48];              b3.y = wb[OUT_CH + 48];

        acc0 = __builtin_amdgcn_wmma_f32_16x16x4_f32(false, a, false, b0,
                                                     (short)0, acc0, false, false);
        acc1 = __builtin_amdgcn_wmma_f32_16x16x4_f32(false, a, false, b1,
                                                     (short)0, acc1, false, false);
        acc2 = __builtin_amdgcn_wmma_f32_16x16x4_f32(false, a, false, b2,
                                                     (short)0, acc2, false, false);
        acc3 = __builtin_amdgcn_wmma_f32_16x16x4_f32(false, a, false, b3,
                                                     (short)0, acc3, false, false);
    }

    // store C/D: vgpr r -> row (tile*16 + r + 8*(lane/16)), col = lane%16
    const int rowhi = (lane >> 4) << 3;  // 0 or 8
    #pragma unroll
    for (int r = 0; r < 8; ++r) {
        int node = tile * 16 + r + rowhi;
        if (node < N) {
            float* hp = h + (size_t)node * OUT_CH + m;
            hp[0]  = acc0[r];
            hp[16] = acc1[r];
            hp[32] = acc2[r];
            hp[48] = acc3[r];
        }
    }
}

// ---- kernel 5: out[col] += norm * h[row], one wave per edge ---------------
__global__ __launch_bounds__(256)
void hgcn_scatter(const long long* __restrict__ ei, const float* __restrict__ cns,
                  const float* __restrict__ dis, const float* __restrict__ h,
                  float* __restrict__ out, int E) {
    int e = blockIdx.x * 8 + (threadIdx.x >> 5);   // 8 waves / 256-thread block
    if (e >= E) return;                             // wave-uniform exit
    const int lane = threadIdx.x & 31;

    long long row = ei[e];
    long long col = ei[(size_t)E + e];
    float norm = dis[row] * dis[col] * expf(cns[e]);

    const int c = lane * 2;                         // 2 channels per lane
    float2 hv = *(const float2*)(h + (size_t)row * OUT_CH + c);
    float* op = out + (size_t)col * OUT_CH + c;
    unsafeAtomicAdd(op,     norm * hv.x);           // native global_atomic_add_f32
    unsafeAtomicAdd(op + 1, norm * hv.y);
}

// ---------------------------------------------------------------------------
extern "C" void kernel_launch(void* const* d_in, const int* in_sizes, int n_in,
                              void* d_out, int out_size, void* d_ws, size_t ws_size,
                              hipStream_t stream) {
    const float*     x    = (const float*)d_in[0];
    const long long* ei   = (const long long*)d_in[1];   // int64 [2, E]
    const float*     cns  = (const float*)d_in[2];
    const float*     W    = (const float*)d_in[3];
    const float*     bias = (const float*)d_in[4];

    const int N = in_sizes[0] / IN_CH;
    const int E = in_sizes[2];

    float* out = (float*)d_out;
    // workspace: [deg_inv_sqrt: N floats (padded to 16B)] [h: N*64 floats]
    float* deg = (float*)d_ws;
    size_t degPad = ((size_t)N + 3) & ~(size_t)3;
    float* h = (float*)d_ws + degPad;

    const int total = N * OUT_CH;
    hgcn_init<<<(total + 255) / 256, 256, 0, stream>>>(out, bias, deg, N);
    hgcn_deg <<<(E + 255) / 256,     256, 0, stream>>>(ei, deg, E);
    hgcn_dis <<<(N + 255) / 256,     256, 0, stream>>>(deg, N);

    const int nTiles = (N + 15) / 16;
    hgcn_gemm_wmma<<<(nTiles + 3) / 4, 128, 0, stream>>>(x, W, h, nTiles, N);

    hgcn_scatter<<<(E + 7) / 8, 256, 0, stream>>>(ei, cns, deg, h, out, E);
}